// GHMC_Loss_38113539784849
// MI455X (gfx1250) — compile-verified
//
#include <hip/hip_runtime.h>

#define ROWS 4096
#define COLS 8192
#define NBINS 30
#define BLOCK 256
#define NWAVES (BLOCK / 32)

// ---------------------------------------------------------------------------
// Kernel 0: zero the global histogram (must be re-zeroed every call; the
// harness does not re-poison d_ws between graph replays).
// ---------------------------------------------------------------------------
__global__ void ghmc_zero_hist(unsigned* __restrict__ hist) {
    if (threadIdx.x < NBINS) hist[threadIdx.x] = 0u;
}

// ---------------------------------------------------------------------------
// Kernel 1: one workgroup per row. Streams logits/target once (float4/int4 ->
// global_load_b128), computes g and stable BCE sharing a single native exp
// per element, accumulates per-wave privatized LDS histograms (ds_add_u32 /
// ds_add_f32), then block-reduces: counts -> global hist (atomic),
// per-row per-bin BCE sums -> S[bin][row] (transposed for coalesced pass 3).
//
// Cheap-math choices (all safe vs harness tolerance):
//  * v_rcp_f32 for 1/(1+e): 1-ulp; used only for bin selection.
//  * __logf(1+e) for log1p(e): 1+e in (1,2] -> no cancellation; abs err ~1e-7.
// ---------------------------------------------------------------------------
__global__ __launch_bounds__(BLOCK) void ghmc_pass1(
    const float* __restrict__ logits,
    const int*   __restrict__ target,
    unsigned*    __restrict__ g_hist,   // [NBINS]
    float*       __restrict__ S)        // [NBINS][ROWS]
{
    __shared__ unsigned s_cnt[NWAVES][NBINS];
    __shared__ float    s_sum[NWAVES][NBINS];

    const int tid  = threadIdx.x;
    const int wave = tid >> 5;

    // zero LDS privatized histograms
    for (int i = tid; i < NWAVES * NBINS; i += BLOCK) {
        (&s_cnt[0][0])[i] = 0u;
        (&s_sum[0][0])[i] = 0.0f;
    }

    const int row = blockIdx.x;
    const float4* lg4 = (const float4*)(logits + (size_t)row * COLS);
    const int4*   tg4 = (const int4*)  (target + (size_t)row * COLS);

    // Warm L2 for a row a few workgroups ahead of this one (one 128B line per
    // lane covers 32KB = one row of logits / one row of target). Speculative
    // prefetch (TH=0): translation failures past the end are silently dropped.
    {
        const char* pfA = (const char*)(logits + ((size_t)row + 8) * COLS);
        const char* pfB = (const char*)(target + ((size_t)row + 8) * COLS);
        __builtin_prefetch(pfA + (size_t)tid * 128, 0, 1);  // global_prefetch_b8
        __builtin_prefetch(pfB + (size_t)tid * 128, 0, 1);
    }

    __syncthreads();

    constexpr int ITERS = COLS / 4 / BLOCK;   // 8
#pragma unroll
    for (int i = 0; i < ITERS; ++i) {
        const int v = i * BLOCK + tid;        // coalesced 16B per lane
        const float4 x4 = lg4[v];
        const int4   t4 = tg4[v];
        const float xs[4] = {x4.x, x4.y, x4.z, x4.w};
        const int   ts[4] = {t4.x, t4.y, t4.z, t4.w};
#pragma unroll
        for (int k = 0; k < 4; ++k) {
            const float x  = xs[k];
            const float tf = (float)ts[k];
            const float e  = __expf(-fabsf(x));              // v_exp_f32 (shared)
            const float inv = __builtin_amdgcn_rcpf(1.0f + e); // sigmoid(|x|), 1 ulp
            const float sml = e * inv;                        // sigmoid(-|x|)
            const bool  pos = (x >= 0.0f);
            const bool  t1  = (ts[k] != 0);
            // g = |sigmoid(x) - t| = sigmoid( t ? -x : x )
            const float g = (pos == t1) ? sml : inv;
            const int b = (int)fminf(g * 29.9999f, 29.0f);    // floor + clamp
            // stable BCE-with-logits: max(x,0) - x*t + log(1+e), 1+e in (1,2]
            const float bce = fmaf(-x, tf, fmaxf(x, 0.0f)) + __logf(1.0f + e);
            atomicAdd(&s_cnt[wave][b], 1u);                   // ds_add_u32
            atomicAdd(&s_sum[wave][b], bce);                  // ds_add_f32
        }
    }
    __syncthreads();

    if (tid < NBINS) {
        unsigned c = 0u;
        float    s = 0.0f;
#pragma unroll
        for (int w = 0; w < NWAVES; ++w) {
            c += s_cnt[w][tid];
            s += s_sum[w][tid];
        }
        atomicAdd(&g_hist[tid], c);                // global_atomic_add_u32
        S[(size_t)tid * ROWS + row] = s;           // transposed store
    }
}

// ---------------------------------------------------------------------------
// Kernel 2: one wave32 computes beta[b] = tot / max(count[b]*nonempty, 1e-4).
// nonempty computed exactly via wave32 ballot over integer counts.
// ---------------------------------------------------------------------------
__global__ void ghmc_beta(const unsigned* __restrict__ hist,
                          float* __restrict__ beta)
{
    const int tid = threadIdx.x;                   // 0..31
    const unsigned c = (tid < NBINS) ? hist[tid] : 0u;
    const unsigned long long mask = __ballot(c > 0u);
    const float nonempty = (float)__popcll(mask);
    if (tid < NBINS) {
        const float tot = (float)ROWS * (float)COLS;   // 2^25, exact in fp32
        const float denom = fmaxf((float)c * nonempty, 1e-4f);
        beta[tid] = tot / denom;
    }
}

// ---------------------------------------------------------------------------
// Kernel 3: out[row] = (1/COLS) * sum_b beta[b] * S[b][row].
// beta cached in LDS; S reads are coalesced (column-major per bin).
// ---------------------------------------------------------------------------
__global__ __launch_bounds__(BLOCK) void ghmc_pass3(
    const float* __restrict__ S,       // [NBINS][ROWS]
    const float* __restrict__ beta,    // [NBINS]
    float*       __restrict__ out)     // [ROWS]
{
    __shared__ float b_sh[NBINS];
    if (threadIdx.x < NBINS) b_sh[threadIdx.x] = beta[threadIdx.x];
    __syncthreads();

    const int row = blockIdx.x * BLOCK + threadIdx.x;
    float acc = 0.0f;
#pragma unroll
    for (int b = 0; b < NBINS; ++b)
        acc = fmaf(b_sh[b], S[(size_t)b * ROWS + row], acc);
    out[row] = acc * (1.0f / (float)COLS);
}

// ---------------------------------------------------------------------------
// Launcher
// ---------------------------------------------------------------------------
extern "C" void kernel_launch(void* const* d_in, const int* in_sizes, int n_in,
                              void* d_out, int out_size, void* d_ws, size_t ws_size,
                              hipStream_t stream)
{
    (void)in_sizes; (void)n_in; (void)out_size; (void)ws_size;

    const float* logits = (const float*)d_in[0];
    const int*   target = (const int*)d_in[1];
    float*       out    = (float*)d_out;

    // Workspace layout (128B-aligned regions):
    //   [0 .. 29]    u32  global histogram
    //   [32 .. 61]   f32  beta
    //   [64 .. ]     f32  S[NBINS][ROWS]
    unsigned* hist = (unsigned*)d_ws;
    float*    beta = (float*)d_ws + 32;
    float*    S    = (float*)d_ws + 64;

    ghmc_zero_hist<<<1, 32, 0, stream>>>(hist);
    ghmc_pass1<<<ROWS, BLOCK, 0, stream>>>(logits, target, hist, S);
    ghmc_beta<<<1, 32, 0, stream>>>(hist, beta);
    ghmc_pass3<<<ROWS / BLOCK, BLOCK, 0, stream>>>(S, beta, out);
}